// Gaussian3D_83648783057476
// MI455X (gfx1250) — compile-verified
//
#include <hip/hip_runtime.h>
#include <stdint.h>

// ---------------------------------------------------------------------------
// cov = R(q_normalized) * diag(exp(log_scale)) * R^T   for N ~ 4M gaussians.
// Memory-bound streaming kernel: 64 B per element, ~11 us floor at 23.3 TB/s.
// WMMA is deliberately NOT used: 3x3-per-item gives <4% tile utilization and
// extra data movement; the CDNA5 feature that *does* help is the async
// global<->LDS DMA path (ASYNCcnt), emitted via inline asm (portable across
// toolchains, per the CDNA5 bridge doc) for the 12B/36B-strided streams.
// ---------------------------------------------------------------------------

#define TPB 256

#if defined(__gfx1250__)
#define USE_ASYNC_LDS 1
// lds addr = low 32 bits of generic pointer (CDNA5 LDS aperture layout)
#define ASYNC_LOAD_B32(ldsptr, gptr)                                   \
    asm volatile("global_load_async_to_lds_b32 %0, %1, off"            \
                 :: "v"((uint32_t)(uintptr_t)(ldsptr)),                \
                    "v"((uint64_t)(uintptr_t)(gptr))                   \
                 : "memory")
#define ASYNC_STORE_B128(gptr, ldsptr)                                 \
    asm volatile("global_store_async_from_lds_b128 %0, %1, off"        \
                 :: "v"((uint64_t)(uintptr_t)(gptr)),                  \
                    "v"((uint32_t)(uintptr_t)(ldsptr))                 \
                 : "memory")
#define WAIT_ASYNC0() asm volatile("s_wait_asynccnt 0" ::: "memory")
#else
#define USE_ASYNC_LDS 0
#endif

__global__ __launch_bounds__(TPB) void Gaussian3D_83648783057476_kernel(
    const float4* __restrict__ quat,   // [n] float4  (w,x,y,z)
    const float*  __restrict__ logs,   // [n*3]
    float*        __restrict__ out,    // [n*9]
    int n)
{
    __shared__ float                 s_ls[3 * TPB];   // staged log_scale
    __shared__ __align__(16) float   s_out[9 * TPB];  // staged covariance rows

    const int tid   = threadIdx.x;
    const int base  = blockIdx.x * TPB;
    const int base3 = base * 3;
    const int base9 = base * 9;

    // ---- stage log_scale (stride-3 rows) into LDS, fully coalesced --------
#if USE_ASYNC_LDS
    #pragma unroll
    for (int k = 0; k < 3; ++k) {
        const int i = tid + k * TPB;
        const int g = base3 + i;
        if (g < 3 * n) {
            ASYNC_LOAD_B32(&s_ls[i], logs + g);      // ASYNCcnt-tracked DMA
        }
    }
#else
    #pragma unroll
    for (int k = 0; k < 3; ++k) {
        const int i = tid + k * TPB;
        const int g = base3 + i;
        if (g < 3 * n) s_ls[i] = logs[g];
    }
#endif

    // ---- per-thread quaternion -> rotation (overlaps the async staging) ---
    const int idx = base + tid;
    float4 q = make_float4(1.f, 0.f, 0.f, 0.f);
    if (idx < n) q = quat[idx];                      // global_load_b128

    const float qn2 = q.x * q.x + q.y * q.y + q.z * q.z + q.w * q.w;
    const float inv = rsqrtf(fmaxf(qn2, 1e-24f));    // == 1/max(|q|,1e-12)
    const float qw = q.x * inv, qx = q.y * inv, qy = q.z * inv, qz = q.w * inv;

    const float r00 = 1.f - 2.f * (qy * qy + qz * qz);
    const float r01 = 2.f * (qx * qy - qz * qw);
    const float r02 = 2.f * (qx * qz + qy * qw);
    const float r10 = 2.f * (qx * qy + qz * qw);
    const float r11 = 1.f - 2.f * (qx * qx + qz * qz);
    const float r12 = 2.f * (qy * qz - qx * qw);
    const float r20 = 2.f * (qx * qz - qy * qw);
    const float r21 = 2.f * (qy * qz + qx * qw);
    const float r22 = 1.f - 2.f * (qx * qx + qy * qy);

#if USE_ASYNC_LDS
    WAIT_ASYNC0();                                   // my wave's DMA done
#endif
    __syncthreads();                                 // everyone's staging done

    // stride-3 LDS read: 3 coprime with 64 banks -> conflict-free
    const float s0 = __expf(s_ls[tid * 3 + 0]);
    const float s1 = __expf(s_ls[tid * 3 + 1]);
    const float s2 = __expf(s_ls[tid * 3 + 2]);

    // cov[i][k] = sum_j R[i][j] * s[j] * R[k][j]  (symmetric: 6 unique)
    const float a0 = s0 * r00, a1 = s1 * r01, a2 = s2 * r02;
    const float b0 = s0 * r10, b1 = s1 * r11, b2 = s2 * r12;
    const float d0 = s0 * r20, d1 = s1 * r21, d2 = s2 * r22;

    const float c00 = a0 * r00 + a1 * r01 + a2 * r02;
    const float c01 = a0 * r10 + a1 * r11 + a2 * r12;
    const float c02 = a0 * r20 + a1 * r21 + a2 * r22;
    const float c11 = b0 * r10 + b1 * r11 + b2 * r12;
    const float c12 = b0 * r20 + b1 * r21 + b2 * r22;
    const float c22 = d0 * r20 + d1 * r21 + d2 * r22;

    // stride-9 LDS write: 9 coprime with 64 banks -> conflict-free
    float* o = &s_out[tid * 9];
    o[0] = c00; o[1] = c01; o[2] = c02;
    o[3] = c01; o[4] = c11; o[5] = c12;
    o[6] = c02; o[7] = c12; o[8] = c22;

    __syncthreads();                                 // s_out fully populated

    // ---- drain 9*TPB floats per block with coalesced b128 traffic ---------
    if (base + TPB <= n) {                           // full tile (common case)
#if USE_ASYNC_LDS
        for (int i = tid; i < (9 * TPB) / 4; i += TPB) {
            ASYNC_STORE_B128(out + base9 + i * 4, &s_out[i * 4]);
        }
        WAIT_ASYNC0();                               // (s_endpgm would also wait)
#else
        const float4* s4 = (const float4*)s_out;
        float4*       g4 = (float4*)(out + base9);
        for (int i = tid; i < (9 * TPB) / 4; i += TPB) g4[i] = s4[i];
#endif
    } else {                                         // ragged tail
        const int vals = (n - base) * 9;
        for (int i = tid; i < vals; i += TPB) out[base9 + i] = s_out[i];
    }
}

extern "C" void kernel_launch(void* const* d_in, const int* in_sizes, int n_in,
                              void* d_out, int out_size, void* d_ws, size_t ws_size,
                              hipStream_t stream) {
    const float4* quat = (const float4*)d_in[0];   // (N,4) float32
    const float*  logs = (const float*)d_in[1];    // (N,3) float32
    float*        out  = (float*)d_out;            // (N,3,3) float32

    const int n      = in_sizes[0] / 4;
    const int blocks = (n + TPB - 1) / TPB;
    Gaussian3D_83648783057476_kernel<<<blocks, TPB, 0, stream>>>(quat, logs, out, n);
}